// RecurrentRGCN_61418032332835
// MI455X (gfx1250) — compile-verified
//
#include <hip/hip_runtime.h>
#include <math.h>

#define NE 100000     // entities
#define NR 460        // num_rels * 2
#define HD 128        // hidden
#define TT 4          // snapshots
#define EE 200000     // edges / snapshot
#define SLOPE ((1.0f/8.0f + 1.0f/3.0f) * 0.5f)

typedef float v2f __attribute__((ext_vector_type(2)));
typedef float v8f __attribute__((ext_vector_type(8)));

__device__ __forceinline__ float sigmoidf(float x) { return 1.0f / (1.0f + expf(-x)); }

// --------------------------------------------------------------------------
// 16xK (A) times Kx128 (B) -> 16x128 accumulated into acc[8] (8 n-tiles).
// Uses V_WMMA_F32_16X16X4_F32. wave32: A 16x4 = 2 VGPR/lane, B 4x16 = 2 VGPR.
// A layout: lane<16 -> row=lane, k=kb,kb+1 ; lane>=16 -> row=lane-16, k=kb+2,kb+3
// B layout: lane<16 -> col=lane, k=kb,kb+1 ; lane>=16 -> col=lane-16, k=kb+2,kb+3
// C/D layout: lane<16 -> M=v, N=lane ; lane>=16 -> M=v+8, N=lane-16
// BT=false: B[k][n] = W[k*ldW + n]   (x @ W)
// BT=true : B[k][n] = W[n*ldW + k]   (x @ W.T)
// --------------------------------------------------------------------------
template<bool BT>
__device__ __forceinline__ void gemm16_acc(const float* __restrict__ aRow,
                                           const float* __restrict__ W,
                                           int ldW, int K, int lane, v8f acc[8]) {
  const int half = lane >> 4;
  const int nl   = lane & 15;
  for (int kb = 0; kb < K; kb += 4) {
    const int kk = kb + half * 2;
    v2f a;
    a.x = aRow[kk];
    a.y = aRow[kk + 1];
#pragma unroll
    for (int nt = 0; nt < 8; ++nt) {
      const int n = nt * 16 + nl;
      v2f b;
      if (BT) { b.x = W[(size_t)n * ldW + kk];  b.y = W[(size_t)n * ldW + kk + 1]; }
      else    { b.x = W[(size_t)kk * ldW + n];  b.y = W[(size_t)(kk + 1) * ldW + n]; }
      acc[nt] = __builtin_amdgcn_wmma_f32_16x16x4_f32(false, a, false, b,
                                                      (short)0, acc[nt], false, false);
    }
  }
}

// ---------------- l2 normalize rows (128 wide), one wave per row ----------
__global__ void l2norm_kernel(const float* __restrict__ in, float* __restrict__ out,
                              int nrows) {
  int row  = blockIdx.x * (blockDim.x >> 5) + (threadIdx.x >> 5);
  int lane = threadIdx.x & 31;
  if (row >= nrows) return;
  const float4* p = (const float4*)(in + (size_t)row * HD);
  float4 v = p[lane];
  float s = v.x * v.x + v.y * v.y + v.z * v.z + v.w * v.w;
#pragma unroll
  for (int off = 16; off > 0; off >>= 1) s += __shfl_xor(s, off, 32);
  float n = sqrtf(s);
  n = n > 1e-12f ? n : 1e-12f;
  float inv = 1.0f / n;
  float4 o; o.x = v.x * inv; o.y = v.y * inv; o.z = v.z * inv; o.w = v.w * inv;
  ((float4*)(out + (size_t)row * HD))[lane] = o;
}

// ---------------- GRUCell on relations: 460 rows, tiny ---------------------
__global__ void gru_rel_kernel(const float* __restrict__ emb_rel,
                               const float* __restrict__ relPrev,
                               const float* __restrict__ Wx,   // (384,256)
                               const float* __restrict__ Wh,   // (384,128)
                               const float* __restrict__ bx,
                               const float* __restrict__ bh,
                               float* __restrict__ relNew) {
  __shared__ float xs[2 * HD];
  __shared__ float hs[HD];
  int r = blockIdx.x, t = threadIdx.x;
  float hv = relPrev[(size_t)r * HD + t];
  xs[t]      = emb_rel[(size_t)r * HD + t];
  xs[t + HD] = hv;
  hs[t]      = hv;
  __syncthreads();
  float xr = bx[t], xz = bx[t + HD], xn = bx[t + 2 * HD];
  for (int k = 0; k < 2 * HD; ++k) {
    float xv = xs[k];
    xr += xv * Wx[(size_t)t * 256 + k];
    xz += xv * Wx[(size_t)(t + HD) * 256 + k];
    xn += xv * Wx[(size_t)(t + 2 * HD) * 256 + k];
  }
  float hr = bh[t], hz = bh[t + HD], hn = bh[t + 2 * HD];
  for (int k = 0; k < HD; ++k) {
    float h2 = hs[k];
    hr += h2 * Wh[(size_t)t * HD + k];
    hz += h2 * Wh[(size_t)(t + HD) * HD + k];
    hn += h2 * Wh[(size_t)(t + 2 * HD) * HD + k];
  }
  float rg = sigmoidf(xr + hr);
  float zg = sigmoidf(xz + hz);
  float ng = tanhf(xn + rg * hn);
  relNew[(size_t)r * HD + t] = (1.0f - zg) * ng + zg * hv;
}

// ---------------- in-degree ------------------------------------------------
__global__ void deg_kernel(const int* __restrict__ dst, float* __restrict__ deg, int E) {
  int e = blockIdx.x * blockDim.x + threadIdx.x;
  if (e < E) unsafeAtomicAdd(&deg[dst[e]], 1.0f);
}

// ---------------- edge message GEMM + scatter ------------------------------
// msg = (h[src] + rel[etype]) @ Wn ; atomically accumulated into agg[dst].
// One wave per 16-edge tile; gathered A tile staged in LDS.
__global__ void edge_msg_kernel(const float* __restrict__ hin,
                                const float* __restrict__ rel,
                                const int* __restrict__ src,
                                const int* __restrict__ dst,
                                const int* __restrict__ etp,
                                const float* __restrict__ Wn,   // (128,128), x@W
                                float* __restrict__ agg,
                                int ntiles) {
  __shared__ float ldsA[4][16 * HD];
  int wave = threadIdx.x >> 5;
  int lane = threadIdx.x & 31;
  int tile = blockIdx.x * 4 + wave;
  if (tile >= ntiles) return;
  int e0 = tile * 16;
  // gather 16 rows of (h[src] + rel[etype]) into this wave's LDS slice
  for (int e = 0; e < 16; ++e) {
    int s = src[e0 + e];
    int q = etp[e0 + e];
    float4 hv = ((const float4*)(hin + (size_t)s * HD))[lane];
    float4 rv = ((const float4*)(rel + (size_t)q * HD))[lane];
    float4 o; o.x = hv.x + rv.x; o.y = hv.y + rv.y; o.z = hv.z + rv.z; o.w = hv.w + rv.w;
    ((float4*)&ldsA[wave][e * HD])[lane] = o;
  }
  v8f acc[8] = {};
  gemm16_acc<false>(&ldsA[wave][(lane & 15) * HD], Wn, HD, HD, lane, acc);
  // scatter-add result rows to agg[dst]
  const int half = lane >> 4, nl = lane & 15;
#pragma unroll
  for (int v = 0; v < 8; ++v) {
    int erow = e0 + half * 8 + v;
    int d = dst[erow];
    float* aggRow = agg + (size_t)d * HD;
#pragma unroll
    for (int nt = 0; nt < 8; ++nt)
      unsafeAtomicAdd(&aggRow[nt * 16 + nl], acc[nt][v]);
  }
}

// ---------------- node self-loop: out = agg/max(deg,1) + h@Wl, rrelu -------
// In place on agg (each wave reads/writes only its own rows).
__global__ void node_selfloop_kernel(const float* __restrict__ hin,
                                     const float* __restrict__ Wl,  // (128,128), x@W
                                     float* __restrict__ aggout,
                                     const float* __restrict__ deg,
                                     int ntiles) {
  int wave = threadIdx.x >> 5;
  int lane = threadIdx.x & 31;
  int tile = blockIdx.x * 4 + wave;
  if (tile >= ntiles) return;
  int r0 = tile * 16;
  v8f acc[8] = {};
  const float* aRow = hin + (size_t)(r0 + (lane & 15)) * HD;
  gemm16_acc<false>(aRow, Wl, HD, HD, lane, acc);
  const int half = lane >> 4, nl = lane & 15;
#pragma unroll
  for (int v = 0; v < 8; ++v) {
    int row = r0 + half * 8 + v;
    float dg = deg[row];
    float inv = 1.0f / (dg > 1.0f ? dg : 1.0f);
#pragma unroll
    for (int nt = 0; nt < 8; ++nt) {
      size_t idx = (size_t)row * HD + nt * 16 + nl;
      float o = aggout[idx] * inv + acc[nt][v];
      aggout[idx] = o >= 0.0f ? o : SLOPE * o;
    }
  }
}

// ---------------- time gate: out = tw*cur + (1-tw)*hprev -------------------
__global__ void timegate_kernel(const float* __restrict__ cur,
                                const float* __restrict__ hprev,
                                const float* __restrict__ Wt,   // (128,128), x@W
                                const float* __restrict__ bt,
                                float* __restrict__ out, int ntiles) {
  int wave = threadIdx.x >> 5;
  int lane = threadIdx.x & 31;
  int tile = blockIdx.x * 4 + wave;
  if (tile >= ntiles) return;
  int r0 = tile * 16;
  v8f acc[8] = {};
  gemm16_acc<false>(cur + (size_t)(r0 + (lane & 15)) * HD, Wt, HD, HD, lane, acc);
  const int half = lane >> 4, nl = lane & 15;
#pragma unroll
  for (int v = 0; v < 8; ++v) {
    int row = r0 + half * 8 + v;
#pragma unroll
    for (int nt = 0; nt < 8; ++nt) {
      int n = nt * 16 + nl;
      size_t idx = (size_t)row * HD + n;
      float tw = sigmoidf(acc[nt][v] + bt[n]);
      out[idx] = tw * cur[idx] + (1.0f - tw) * hprev[idx];
    }
  }
}

// ---------------- gated fusion: gate=sig([h1|hp]@Wg.T+b) -------------------
__global__ void gatefusion_kernel(const float* __restrict__ h1,
                                  const float* __restrict__ hprev,
                                  const float* __restrict__ Wg,  // (128,256), x@W.T
                                  const float* __restrict__ bg,
                                  float* __restrict__ out, int ntiles) {
  int wave = threadIdx.x >> 5;
  int lane = threadIdx.x & 31;
  int tile = blockIdx.x * 4 + wave;
  if (tile >= ntiles) return;
  int r0 = tile * 16;
  v8f acc[8] = {};
  gemm16_acc<true>(h1    + (size_t)(r0 + (lane & 15)) * HD, Wg,      256, HD, lane, acc);
  gemm16_acc<true>(hprev + (size_t)(r0 + (lane & 15)) * HD, Wg + HD, 256, HD, lane, acc);
  const int half = lane >> 4, nl = lane & 15;
#pragma unroll
  for (int v = 0; v < 8; ++v) {
    int row = r0 + half * 8 + v;
#pragma unroll
    for (int nt = 0; nt < 8; ++nt) {
      int n = nt * 16 + nl;
      size_t idx = (size_t)row * HD + n;
      float g = sigmoidf(acc[nt][v] + bg[n]);
      out[idx] = g * h1[idx] + (1.0f - g) * hprev[idx];
    }
  }
}

// ---------------- temporal diff: out = h2 + relu((h2-hp)@Wd.T + b) ---------
__global__ void tdiff_kernel(const float* __restrict__ h2,
                             const float* __restrict__ hprev,
                             const float* __restrict__ Wd,   // (128,128), x@W.T
                             const float* __restrict__ bd,
                             float* __restrict__ out, int ntiles) {
  __shared__ float ldsD[4][16 * HD];
  int wave = threadIdx.x >> 5;
  int lane = threadIdx.x & 31;
  int tile = blockIdx.x * 4 + wave;
  if (tile >= ntiles) return;
  int r0 = tile * 16;
  for (int e = 0; e < 16; ++e) {
    int row = r0 + e;
    float4 a = ((const float4*)(h2    + (size_t)row * HD))[lane];
    float4 b = ((const float4*)(hprev + (size_t)row * HD))[lane];
    float4 o; o.x = a.x - b.x; o.y = a.y - b.y; o.z = a.z - b.z; o.w = a.w - b.w;
    ((float4*)&ldsD[wave][e * HD])[lane] = o;
  }
  v8f acc[8] = {};
  gemm16_acc<true>(&ldsD[wave][(lane & 15) * HD], Wd, HD, HD, lane, acc);
  const int half = lane >> 4, nl = lane & 15;
#pragma unroll
  for (int v = 0; v < 8; ++v) {
    int row = r0 + half * 8 + v;
#pragma unroll
    for (int nt = 0; nt < 8; ++nt) {
      int n = nt * 16 + nl;
      size_t idx = (size_t)row * HD + n;
      float rl = acc[nt][v] + bd[n];
      out[idx] = h2[idx] + (rl > 0.0f ? rl : 0.0f);
    }
  }
}

// ===========================================================================
extern "C" void kernel_launch(void* const* d_in, const int* in_sizes, int n_in,
                              void* d_out, int out_size, void* d_ws, size_t ws_size,
                              hipStream_t stream) {
  const int*   src     = (const int*)d_in[0];
  const int*   dst     = (const int*)d_in[1];
  const int*   etp     = (const int*)d_in[2];
  const float* demb    = (const float*)d_in[3];
  const float* emb_rel = (const float*)d_in[4];
  const float* Wn1 = (const float*)d_in[5],  *Wl1 = (const float*)d_in[6];
  const float* Wn2 = (const float*)d_in[7],  *Wl2 = (const float*)d_in[8];
  const float* gWx = (const float*)d_in[9],  *gWh = (const float*)d_in[10];
  const float* gbx = (const float*)d_in[11], *gbh = (const float*)d_in[12];
  const float* gateW = (const float*)d_in[13], *gateB = (const float*)d_in[14];
  const float* tdW   = (const float*)d_in[15], *tdB   = (const float*)d_in[16];
  const float* tgW   = (const float*)d_in[17], *tgB   = (const float*)d_in[18];

  float* w    = (float*)d_ws;
  float* bufA = w; w += (size_t)NE * HD;
  float* bufB = w; w += (size_t)NE * HD;
  float* bufC = w; w += (size_t)NE * HD;
  float* deg  = w; w += NE;
  float* relA = w; w += (size_t)NR * HD;
  float* relB = w; w += (size_t)NR * HD;
  (void)relB; (void)in_sizes; (void)n_in; (void)out_size; (void)ws_size;

  const int ETILES = EE / 16;   // 12500
  const int NTILES = NE / 16;   // 6250
  const int egrid  = ETILES / 4;         // 3125
  const int ngrid  = (NTILES + 3) / 4;   // 1563
  dim3 blk(128);

  // h0 = l2norm(dynamic_emb)
  l2norm_kernel<<<NE / 8, 256, 0, stream>>>(demb, bufA, NE);

  float* P = bufA;   // h_prev carry
  float* X = bufB;   // scratch 1
  float* Y = bufC;   // scratch 2
  const float* relPrev = emb_rel;

  for (int t = 0; t < TT; ++t) {
    const int* s = src + (size_t)t * EE;
    const int* d = dst + (size_t)t * EE;
    const int* e = etp + (size_t)t * EE;

    float* relNew = (t & 1) ? relB : relA;
    gru_rel_kernel<<<NR, HD, 0, stream>>>(emb_rel, relPrev, gWx, gWh, gbx, gbh, relNew);
    relPrev = relNew;

    hipMemsetAsync(deg, 0, NE * sizeof(float), stream);
    deg_kernel<<<(EE + 255) / 256, 256, 0, stream>>>(d, deg, EE);

    // layer 1: in=P, agg/out=X
    hipMemsetAsync(X, 0, (size_t)NE * HD * sizeof(float), stream);
    edge_msg_kernel<<<egrid, blk, 0, stream>>>(P, relNew, s, d, e, Wn1, X, ETILES);
    node_selfloop_kernel<<<ngrid, blk, 0, stream>>>(P, Wl1, X, deg, NTILES);

    // layer 2: in=X, agg/out=Y
    hipMemsetAsync(Y, 0, (size_t)NE * HD * sizeof(float), stream);
    edge_msg_kernel<<<egrid, blk, 0, stream>>>(X, relNew, s, d, e, Wn2, Y, ETILES);
    node_selfloop_kernel<<<ngrid, blk, 0, stream>>>(X, Wl2, Y, deg, NTILES);

    // cur = l2norm (in place on Y)
    l2norm_kernel<<<NE / 8, 256, 0, stream>>>(Y, Y, NE);

    // time gate: h1 -> X   (layer1 contents dead)
    timegate_kernel<<<ngrid, blk, 0, stream>>>(Y, P, tgW, tgB, X, NTILES);
    // gated fusion: h2 -> Y (cur dead)
    gatefusion_kernel<<<ngrid, blk, 0, stream>>>(X, P, gateW, gateB, Y, NTILES);
    // temporal diff: h3 -> X (or final output)
    float* outP = (t == TT - 1) ? (float*)d_out : X;
    tdiff_kernel<<<ngrid, blk, 0, stream>>>(Y, P, tdW, tdB, outP, NTILES);

    // rotate buffers: new carry = outP; freed carry becomes scratch
    float* oldP = P;
    P = outP;
    X = oldP;
    // Y stays as scratch 2
  }
}